// NURBSSurface_80625126080689
// MI455X (gfx1250) — compile-verified
//
#include <hip/hip_runtime.h>

#define OUT_X 1024
#define OUT_Y 1024
#define NC    32          // control points per axis
#define DEG   3
#define NK    36          // knots per axis
#define EPSV  1e-5f

typedef __attribute__((ext_vector_type(2))) float v2f;
typedef __attribute__((ext_vector_type(8))) float v8f;

// ---------------------------------------------------------------------------
// evaluation grid: jnp.linspace(EPS, 1-EPS, N)
__device__ __forceinline__ float ev_at(int t, int n) {
  return EPSV + (float)t * ((1.0f - 2.0f * EPSV) / (float)(n - 1));
}

// span search + Cox-de-Boor basis, replicating the reference exactly
// (argmin-first-occurrence span; denom==0 -> temp=0.0001 quirk)
__device__ __forceinline__ void span_basis(const float* kv, float ev,
                                           int& span_out, float basis[DEG + 1]) {
  int span = DEG;
  float best = 3.4e38f;
  #pragma unroll
  for (int i = 0; i < NK - 2 * DEG; ++i) {   // internal knots kv[3..32]
    float diff = ev - kv[DEG + i];
    diff = (diff > 1e-8f) ? diff : 1.0f;
    if (diff < best) { best = diff; span = i + DEG; }
  }
  basis[0] = 1.0f; basis[1] = 0.0f; basis[2] = 0.0f; basis[3] = 0.0f;
  #pragma unroll
  for (int k = 1; k <= DEG; ++k) {
    float saved = 0.0f;
    #pragma unroll
    for (int r = 0; r < DEG; ++r) {
      if (r >= k) break;
      float left  = kv[span + r + 1];
      float right = kv[span + 1 - k + r];
      float denom = (left - ev) + (ev - right);
      float temp  = (denom == 0.0f) ? 0.0001f : (basis[r] / denom);
      basis[r] = saved + (left - ev) * temp;
      saved    = (ev - right) * temp;
    }
    basis[k] = saved;
  }
  span_out = span;
}

// ---------------------------------------------------------------------------
// Kernel A: knot normalization (cumsum), spans, basis.
// blocks 0-3: x axis -> bx[4][1024], sx[1024]
// blocks 4-7: y axis -> dense Wy[1024][32] (basis scattered by span)
__global__ void nurbs_prep(const float* __restrict__ knots_x,
                           const float* __restrict__ knots_y,
                           float* __restrict__ bx, int* __restrict__ sx,
                           float* __restrict__ wy) {
  __shared__ float kv[NK];
  const bool isY = (blockIdx.x >= 4);
  const float* kraw = isY ? knots_y : knots_x;

  if (threadIdx.x == 0) {
    float c = 0.0f;
    for (int j = 0; j < NK; ++j) {
      float v = kraw[j];
      v = (v < 0.0f) ? 1e-4f : v;       // where(kv<0, 1e-4, kv)
      c += v;                            // cumsum
      kv[j] = c;
    }
    float c0 = kv[0];
    float dd = kv[NK - 1] - c0;
    for (int j = 0; j < NK; ++j) kv[j] = (kv[j] - c0) / dd;
  }
  __syncthreads();

  int t = (blockIdx.x & 3) * blockDim.x + threadIdx.x;   // 0..1023
  float ev = ev_at(t, isY ? OUT_Y : OUT_X);
  int span; float b[DEG + 1];
  span_basis(kv, ev, span, b);

  if (!isY) {
    sx[t] = span;
    #pragma unroll
    for (int l = 0; l <= DEG; ++l) bx[l * OUT_X + t] = b[l];
  } else {
    float w[NC];
    #pragma unroll
    for (int i = 0; i < NC; ++i) w[i] = 0.0f;
    #pragma unroll
    for (int l = 0; l <= DEG; ++l) {
      int i = span - DEG - l;            // idx = span - (3+l), wrap
      if (i < 0) i += NC;
      w[i] = b[l];
    }
    #pragma unroll
    for (int i = 0; i < NC; ++i) wy[t * NC + i] = w[i];
  }
}

// ---------------------------------------------------------------------------
// Kernel B: stage 1 (sparse form): T[d][x][j] = sum_l bx[l][x] * cp[idx(x,l)][j][d]
__global__ void nurbs_stage1(const float* __restrict__ cp,
                             const float* __restrict__ bx,
                             const int* __restrict__ sx,
                             float* __restrict__ T) {
  int gid = blockIdx.x * blockDim.x + threadIdx.x;   // 0..32767
  int x = gid >> 5;
  int j = gid & 31;
  int span = sx[x];
  float a0 = 0.0f, a1 = 0.0f, a2 = 0.0f;
  #pragma unroll
  for (int l = 0; l <= DEG; ++l) {
    int i = span - DEG - l;
    if (i < 0) i += NC;
    float w = bx[l * OUT_X + x];
    const float* p = cp + (i * NC + j) * 3;
    a0 += w * p[0];
    a1 += w * p[1];
    a2 += w * p[2];
  }
  T[0 * (OUT_X * NC) + gid] = a0;
  T[1 * (OUT_X * NC) + gid] = a1;
  T[2 * (OUT_X * NC) + gid] = a2;
}

// ---------------------------------------------------------------------------
// Kernel C: stage 2 with V_WMMA_F32_16X16X4_F32.
// surf_d = T_d (1024x32) * Wy^T (32x1024). One wave owns one 16x16 (x,y) tile
// and all 3 d-planes; 8 K-steps of K=4 cover K=32. B fragments (Wy) are reused
// across the 3 d accumulations.
__global__ void __launch_bounds__(256)
nurbs_stage2(const float* __restrict__ T, const float* __restrict__ wy,
             float* __restrict__ out) {
  int wave = threadIdx.x >> 5;
  int lane = threadIdx.x & 31;
  int tile = blockIdx.x * 8 + wave;     // 0..4095 (64 x-tiles * 64 y-tiles)
  int tx = tile >> 6;
  int ty = tile & 63;
  int x0 = tx << 4;
  int y0 = ty << 4;

  int row  = lane & 15;                 // M for A, N for B (mirrored layouts)
  int koff = (lane >> 4) << 1;          // lanes 0-15: K={0,1}; 16-31: K={2,3}

  // B fragments: B[k][n] = Wy[y0+n][k]  -> same row-major float2 pattern as A
  v2f bfrag[8];
  const float* wrow = wy + (y0 + row) * NC + koff;
  #pragma unroll
  for (int kk = 0; kk < 8; ++kk)
    bfrag[kk] = *(const v2f*)(wrow + kk * 4);

  v8f acc[3];
  #pragma unroll
  for (int d = 0; d < 3; ++d) {
    const float* arow = T + d * (OUT_X * NC) + (x0 + row) * NC + koff;
    v8f c = {};
    #pragma unroll
    for (int kk = 0; kk < 8; ++kk) {
      v2f a = *(const v2f*)(arow + kk * 4);
      // (neg_a, A, neg_b, B, c_mod, C, reuse_a, reuse_b)
      c = __builtin_amdgcn_wmma_f32_16x16x4_f32(
          false, a, false, bfrag[kk], (short)0, c, false, false);
    }
    acc[d] = c;
  }

  // D layout: VGPR v, lanes 0-15: M=v, N=lane; lanes 16-31: M=v+8, N=lane-16
  int mbase = (lane >> 4) << 3;
  #pragma unroll
  for (int v = 0; v < 8; ++v) {
    int gx = x0 + mbase + v;
    int gy = y0 + row;
    float* p = out + (((gx << 10) + gy) * 3);  // (x,y,d) contiguous in d
    p[0] = acc[0][v];
    p[1] = acc[1][v];
    p[2] = acc[2][v];
  }
}

// ---------------------------------------------------------------------------
extern "C" void kernel_launch(void* const* d_in, const int* in_sizes, int n_in,
                              void* d_out, int out_size, void* d_ws, size_t ws_size,
                              hipStream_t stream) {
  const float* cp = (const float*)d_in[0];  // (32,32,3)
  const float* kx = (const float*)d_in[1];  // (1,36)
  const float* ky = (const float*)d_in[2];  // (1,36)
  float* out = (float*)d_out;               // (1,1024,1024,3)

  float* ws = (float*)d_ws;
  float* bx = ws;                  // 4*1024
  int*   sx = (int*)(ws + 4096);   // 1024
  float* wy = ws + 5120;           // 1024*32
  float* T  = ws + 37888;          // 3*1024*32  (ends at 136192 floats ~ 545 KB)

  nurbs_prep  <<<  8, 256, 0, stream>>>(kx, ky, bx, sx, wy);
  nurbs_stage1<<<128, 256, 0, stream>>>(cp, bx, sx, T);
  nurbs_stage2<<<512, 256, 0, stream>>>(T, wy, out);
}